// AtomEmbedding_82154134438740
// MI455X (gfx1250) — compile-verified
//
#include <hip/hip_runtime.h>

typedef __attribute__((ext_vector_type(2))) float v2f;
typedef __attribute__((ext_vector_type(8))) float v8f;

#define HAVE_WMMA_F32X4 __has_builtin(__builtin_amdgcn_wmma_f32_16x16x4_f32)

__device__ __forceinline__ int clampi(int x, int lo, int hi) {
    return x < lo ? lo : (x > hi ? hi : x);
}
__device__ __forceinline__ float4 ld4(const float* p) { return *(const float4*)p; }
__device__ __forceinline__ float2 ld2(const float* p) { return *(const float2*)p; }

__global__ __launch_bounds__(128) void atom_embed_kernel(
    const float* __restrict__ ai,      // [n,78]
    const int*   __restrict__ lut,     // [lutN]
    const int*   __restrict__ ringv,   // [7]
    const float* __restrict__ E_elem,  // [7,4]
    const float* __restrict__ E_deg,   // [7,4]
    const float* __restrict__ E_ring,  // [2,4]
    const float* __restrict__ E_chg,   // [8,4]
    const float* __restrict__ E_arom,  // [2,4]
    const float* __restrict__ E_hyb,   // [6,4]
    const float* __restrict__ E_hyd,   // [5,4]
    const float* __restrict__ T_func,  // [18,2,2]
    const float* __restrict__ E_don,   // [2,2]
    const float* __restrict__ E_acc,   // [2,2]
    const float* __restrict__ E_rs,    // [7,4]
    const float* __restrict__ E_an,    // [5,4]
    const float* __restrict__ E_fu,    // [8,4]
    const float* __restrict__ W36,     // [4,36]
    const float* __restrict__ B4,      // [4]
    const float* __restrict__ W48,     // [16,48]
    const float* __restrict__ B16,     // [16]
    float* __restrict__ out,           // [n,64]
    int n, int lutN)
{
    __shared__ float sC[18][2][4];   // flag-group contribution vectors
    __shared__ float sB4[4];
    __shared__ int   sRV[7];

    const int t = threadIdx.x;
    if (t < 144) {
        int j = t >> 3, v = (t >> 2) & 1, o = t & 3;
        sC[j][v][o] = T_func[j * 4 + v * 2 + 0] * W36[o * 36 + 2 * j]
                    + T_func[j * 4 + v * 2 + 1] * W36[o * 36 + 2 * j + 1];
    }
    if (t < 4) sB4[t] = B4[t];
    if (t < 7) sRV[t] = ringv[t];
    __syncthreads();

    // ---------------- per-atom embedding half: output cols 0..47 ----------------
    const long long a = (long long)blockIdx.x * 128 + t;
    if (a < n) {
        // cols 0..29 as 15 aligned b64 loads (row base is 8B aligned: 78 even)
        const float2* rp = (const float2*)(ai + (size_t)a * 78);
        float r[30];
        #pragma unroll
        for (int i = 0; i < 15; ++i) { float2 v = rp[i]; r[2*i] = v.x; r[2*i+1] = v.y; }

        // element index via LUT with out-of-range Z -> 0
        int zr = (int)r[0];
        int z  = (zr >= 0 && zr < lutN) ? zr : 0;
        int i0 = clampi(lut[z], 0, 6);

        float4 x0 = ld4(E_elem + i0 * 4);
        float4 x1 = ld4(E_deg  + clampi((int)r[1], 0, 6) * 4);
        float4 x2 = ld4(E_ring + clampi((int)r[5] + 1, 0, 1) * 4);
        float4 x3 = ld4(E_chg  + clampi((int)r[2], 0, 7) * 4);
        float4 x4 = ld4(E_arom + clampi((int)r[4], 0, 1) * 4);
        float4 x5 = ld4(E_hyb  + clampi((int)r[3], 0, 5) * 4);
        float4 x6 = ld4(E_hyd  + clampi((int)r[6], 0, 4) * 4);

        // flags4: bias + sum of 18 precomputed 4-vectors
        float f0 = sB4[0], f1 = sB4[1], f2 = sB4[2], f3 = sB4[3];
        #pragma unroll
        for (int j = 0; j < 18; ++j) {
            int v = clampi((int)r[7 + j], 0, 1);
            const float* cp = sC[j][v];
            f0 += cp[0]; f1 += cp[1]; f2 += cp[2]; f3 += cp[3];
        }

        float2 x25 = ld2(E_don + clampi((int)r[25], 0, 1) * 2);
        float2 x26 = ld2(E_acc + clampi((int)r[26], 0, 1) * 2);

        // ring-size remap {0,3,4,5,6,7,8} -> {0..6}, unknown -> 6 (first match wins)
        int raw27 = (int)r[27];
        int m27 = 6;
        #pragma unroll
        for (int i = 6; i >= 0; --i) if (sRV[i] == raw27) m27 = i;
        float4 x27 = ld4(E_rs + m27 * 4);
        float4 x28 = ld4(E_an + clampi((int)r[28], 0, 4) * 4);
        float4 x29 = ld4(E_fu + clampi((int)r[29], 0, 7) * 4);

        float4* op = (float4*)(out + (size_t)a * 64);   // 256B-aligned row
        op[0]  = x0;  op[1] = x1;  op[2] = x2;  op[3] = x3;
        op[4]  = x4;  op[5] = x5;  op[6] = x6;
        op[7]  = make_float4(f0, f1, f2, f3);
        op[8]  = make_float4(x25.x, x25.y, x26.x, x26.y);
        op[9]  = x27; op[10] = x28; op[11] = x29;
    }

    // ---------------- WMMA half: bond_env (48 -> 16), output cols 48..63 --------
    const int lane = t & 31;
    const int wave = t >> 5;        // 0..3
    const int hi   = lane >> 4;     // half-wave select (K/row-offset split)
    const int nch  = lane & 15;     // output channel (N) / A-row (M)

#if HAVE_WMMA_F32X4
    const float bias = B16[nch];
    #pragma unroll
    for (int tt = 0; tt < 2; ++tt) {
        const long long tileBase = (long long)blockIdx.x * 128 + (wave * 2 + tt) * 16;
        if (tileBase >= n) continue;              // wave-uniform skip
        long long arow = tileBase + nch;
        if (arow >= n) arow = n - 1;              // clamp keeps EXEC all-ones

        v8f c;
        #pragma unroll
        for (int q = 0; q < 8; ++q) c[q] = bias;

        const float* abase = ai  + (size_t)arow * 78 + 30 + 2 * hi;  // bond cols
        const float* wbase = W48 + (size_t)nch  * 48 + 2 * hi;       // B = W^T
        #pragma unroll
        for (int k0 = 0; k0 < 48; k0 += 4) {
            float2 at = ld2(abase + k0);
            float2 wt = ld2(wbase + k0);
            v2f af = {at.x, at.y};
            v2f bf = {wt.x, wt.y};
            // D(16x16,f32) += A(16x4,f32) * B(4x16,f32)
            c = __builtin_amdgcn_wmma_f32_16x16x4_f32(
                    false, af, false, bf, (short)0, c, false, false);
        }

        // One base pointer; 8 stores with immediate offsets (q*64 floats).
        float* obase = out + (size_t)(tileBase + 8 * hi) * 64 + 48 + nch;
        if (tileBase + 16 <= n) {
            // full tile (always the case when n % 16 == 0): straight-line stores
            #pragma unroll
            for (int q = 0; q < 8; ++q) obase[(size_t)q * 64] = c[q];
        } else {
            #pragma unroll
            for (int q = 0; q < 8; ++q)
                if (tileBase + q + 8 * hi < n) obase[(size_t)q * 64] = c[q];
        }
    }
#else
    // Scalar fallback (only if the f32 WMMA builtin is unavailable)
    if (a < n) {
        const float* bp = ai + (size_t)a * 78 + 30;
        #pragma unroll 4
        for (int cch = 0; cch < 16; ++cch) {
            float acc = B16[cch];
            const float* wp = W48 + cch * 48;
            #pragma unroll
            for (int k = 0; k < 48; ++k) acc = fmaf(bp[k], wp[k], acc);
            out[(size_t)a * 64 + 48 + cch] = acc;
        }
    }
#endif
}

extern "C" void kernel_launch(void* const* d_in, const int* in_sizes, int n_in,
                              void* d_out, int out_size, void* d_ws, size_t ws_size,
                              hipStream_t stream) {
    const int n    = in_sizes[0] / 78;
    const int lutN = in_sizes[1];
    const int blocks = (n + 127) / 128;
    atom_embed_kernel<<<blocks > 0 ? blocks : 1, 128, 0, stream>>>(
        (const float*)d_in[0],  (const int*)d_in[1],   (const int*)d_in[2],
        (const float*)d_in[3],  (const float*)d_in[4], (const float*)d_in[5],
        (const float*)d_in[6],  (const float*)d_in[7], (const float*)d_in[8],
        (const float*)d_in[9],  (const float*)d_in[10],(const float*)d_in[11],
        (const float*)d_in[12], (const float*)d_in[13],(const float*)d_in[14],
        (const float*)d_in[15], (const float*)d_in[16],(const float*)d_in[17],
        (const float*)d_in[18], (const float*)d_in[19],
        (float*)d_out, n, lutN);
}